// Unpooling_56427280335301
// MI455X (gfx1250) — compile-verified
//
#include <hip/hip_runtime.h>

// unsorted_segment_max scatter for MI455X (gfx1250, wave32).
//
// Strategy:
//  - Output (102.8 MB) fits in the 192 MB global L2 -> do the scatter as
//    L2-resident non-returning i32 atomics (global_atomic_max_i32).
//  - For v >= 0.0f, float ordering == signed-int ordering of the bit pattern,
//    and the reference maps any negative segment max to -inf anyway. So:
//    init out to bits(-inf)=0xFF800000 (negative as int), atomic-max only the
//    non-negative values' bit patterns. Empty / all-negative segments stay
//    -inf with NO post-pass, and ~half the atomics are filtered out.
//  - Inputs are streamed exactly once: load them with non-temporal hints
//    (global_load_b128 th:TH_LOAD_NT) so they don't evict the output from L2.

typedef __attribute__((ext_vector_type(4))) float v4f;
typedef __attribute__((ext_vector_type(4))) int   v4i;

#define NEG_INF_BITS 0xFF800000

// ---- Kernel 1: initialize output to -inf, 16B per lane ---------------------
__global__ void __launch_bounds__(256) seg_init_kernel(v4i* __restrict__ out,
                                                       int nvec) {
    int i = blockIdx.x * blockDim.x + threadIdx.x;
    if (i < nvec) {
        v4i v;
        v.x = (int)NEG_INF_BITS;
        v.y = (int)NEG_INF_BITS;
        v.z = (int)NEG_INF_BITS;
        v.w = (int)NEG_INF_BITS;
        out[i] = v;  // global_store_b128
    }
}

// ---- Kernel 2: vectorized scatter-max --------------------------------------
__device__ __forceinline__ void scatter_one(int* __restrict__ out,
                                            float v, int idx, int num_seg) {
    // Reference applies where(out < 0, -inf, out): negative maxima are
    // indistinguishable from empty segments, so skip negative values.
    if (v >= 0.0f) {
        int bits = __float_as_int(v);
        if (bits < 0) bits = 0;  // flush -0.0 -> +0.0 so int order == float order
        if ((unsigned)idx < (unsigned)num_seg) {
            atomicMax(out + idx, bits);  // global_atomic_max_i32 (no return)
        }
    }
}

__global__ void __launch_bounds__(256) seg_max_kernel(
    const v4f* __restrict__ vals,
    const v4i* __restrict__ idxs,
    int* __restrict__ out,
    int nvec, int num_seg) {
    int i = blockIdx.x * blockDim.x + threadIdx.x;
    if (i < nvec) {
        // Non-temporal b128 loads: streamed once, keep L2 for the output.
        v4f v  = __builtin_nontemporal_load(vals + i);
        v4i ix = __builtin_nontemporal_load(idxs + i);
        scatter_one(out, v.x, ix.x, num_seg);
        scatter_one(out, v.y, ix.y, num_seg);
        scatter_one(out, v.z, ix.z, num_seg);
        scatter_one(out, v.w, ix.w, num_seg);
    }
}

extern "C" void kernel_launch(void* const* d_in, const int* in_sizes, int n_in,
                              void* d_out, int out_size, void* d_ws, size_t ws_size,
                              hipStream_t stream) {
    const v4f* layer   = (const v4f*)d_in[0];   // float32 (32,56,56,64)
    const v4i* indices = (const v4i*)d_in[1];   // int32   (32,56,56,64)
    int* out = (int*)d_out;                     // float32 (32,112,112,64), as bits

    const int n_in_elems = in_sizes[0];         // 6,422,528 (multiple of 4)
    const int n_out      = out_size;            // 25,690,112 (multiple of 4)

    // Phase 1: out = -inf everywhere (b128 stores, exact fit).
    {
        int nvec   = n_out / 4;                 // 6,422,528
        int block  = 256;
        int grid   = (nvec + block - 1) / block;
        seg_init_kernel<<<grid, block, 0, stream>>>((v4i*)out, nvec);
    }

    // Phase 2: atomic scatter-max of non-negative values (same stream ->
    // ordered after init).
    {
        int nvec   = n_in_elems / 4;            // 1,605,632
        int block  = 256;
        int grid   = (nvec + block - 1) / block;
        seg_max_kernel<<<grid, block, 0, stream>>>(layer, indices, out,
                                                   nvec, n_out);
    }
}